// FragNetFineTuneBase_21878563406403
// MI455X (gfx1250) — compile-verified
//
#include <hip/hip_runtime.h>
#include <hip/hip_bf16.h>

typedef __attribute__((ext_vector_type(2))) float v2f;
typedef __attribute__((ext_vector_type(8))) float v8f;

__device__ __forceinline__ int lower_bound_i32(const int* __restrict__ a, int n, int v) {
    int lo = 0, hi = n;
    while (lo < hi) {
        int mid = (lo + hi) >> 1;
        if (a[mid] < v) lo = mid + 1; else hi = mid;
    }
    return lo;
}

// One block = 16 consecutive graphs (output rows); 8 waves x 16 embedding cols each.
// Segment-sum as indicator-matrix WMMA (exact fp32):
//   C[16 graphs x 16 cols] += A[16 graphs x 4 rows] * B[4 rows x 16 cols]
// via V_WMMA_F32_16X16X4_F32. Sorted ids => block owns its 16 output rows, no atomics.
// Main loop is branch-free; tail uses clamped loads so EXEC stays all-1s (WMMA requirement).
// Data loads are non-temporal: 1.54 GB streamed once, zero reuse, don't churn the 192MB L2.
__global__ __launch_bounds__(256, 8)
void frag_pool_wmma_kernel(const float* __restrict__ xa, const float* __restrict__ xf,
                           const int* __restrict__ ba, const int* __restrict__ bf,
                           int na, int nf, float* __restrict__ out)
{
    const int isFrag = blockIdx.y;
    const float* __restrict__ src = isFrag ? xf : xa;
    const int*   __restrict__ idx = isFrag ? bf : ba;
    const int n      = isFrag ? nf : na;
    const int colOff = isFrag ? 128 : 0;

    const int segBase = (int)blockIdx.x << 4;          // first of the 16 graphs this block owns
    const int start = lower_bound_i32(idx, n, segBase);
    const int end   = lower_bound_i32(idx, n, segBase + 16);

    const int lane = threadIdx.x & 31;
    const int wave = threadIdx.x >> 5;
    const int half = lane >> 4;            // 0: lanes 0-15, 1: lanes 16-31
    const int mn   = lane & 15;            // M index for A-layout, N index for B/C-layout
    const int kb   = half << 1;            // per-lane K base in A/B layout: {0,1} or {2,3}
    const int col  = (wave << 4) + mn;     // embedding column 0..127

    v8f c0 = {};
    v8f c1 = {};

    const int total = end - start;
    const int nFull = total & ~7;          // rows handled by the unguarded main loop

    // Lane-resident marching pointers: kb and col folded in once; loads below use
    // constant immediate offsets only.
    const float* p = src + (size_t)(start + kb) * 128 + col;
    const int*   q = idx + start + kb;

    for (int t = 0; t < nFull; t += 8) {
        __builtin_prefetch(p + 40 * 128, 0, 1);        // stay ~5 iterations ahead of HBM

        // all 8 rows of this iteration are < end: no guards anywhere.
        // NT hint: streamed once, bypass temporal caching.
        float b00 = __builtin_nontemporal_load(p);
        float b01 = __builtin_nontemporal_load(p + 128);
        float b10 = __builtin_nontemporal_load(p + 4 * 128);
        float b11 = __builtin_nontemporal_load(p + 5 * 128);
        int   i00 = q[0];
        int   i01 = q[1];
        int   i10 = q[4];
        int   i11 = q[5];

        v2f a0 = { (i00 - segBase == mn) ? 1.0f : 0.0f,
                   (i01 - segBase == mn) ? 1.0f : 0.0f };
        v2f b0 = { b00, b01 };
        c0 = __builtin_amdgcn_wmma_f32_16x16x4_f32(
                false, a0, false, b0, (short)0, c0, false, false);

        v2f a1 = { (i10 - segBase == mn) ? 1.0f : 0.0f,
                   (i11 - segBase == mn) ? 1.0f : 0.0f };
        v2f b1 = { b10, b11 };
        c1 = __builtin_amdgcn_wmma_f32_16x16x4_f32(
                false, a1, false, b1, (short)0, c1, false, false);

        p += 8 * 128;
        q += 8;
    }

    // Tail (0..7 rows). Block-uniform branch: EXEC stays all-1s for the WMMAs.
    // Out-of-range rows are CLAMPED (always-in-bounds loads) and killed via a 0 indicator.
    const int base = start + nFull;
    if (base < end) {
        const int last = end - 1;
        int r00 = base + kb, r01 = r00 + 1, r10 = r00 + 4, r11 = r00 + 5;
        int m00 = min(r00, last), m01 = min(r01, last);
        int m10 = min(r10, last), m11 = min(r11, last);

        float b00 = __builtin_nontemporal_load(src + (size_t)m00 * 128 + col);
        float b01 = __builtin_nontemporal_load(src + (size_t)m01 * 128 + col);
        float b10 = __builtin_nontemporal_load(src + (size_t)m10 * 128 + col);
        float b11 = __builtin_nontemporal_load(src + (size_t)m11 * 128 + col);
        int   i00 = idx[m00];
        int   i01 = idx[m01];
        int   i10 = idx[m10];
        int   i11 = idx[m11];

        v2f a0 = { (r00 < end && i00 - segBase == mn) ? 1.0f : 0.0f,
                   (r01 < end && i01 - segBase == mn) ? 1.0f : 0.0f };
        v2f b0 = { b00, b01 };
        c0 = __builtin_amdgcn_wmma_f32_16x16x4_f32(
                false, a0, false, b0, (short)0, c0, false, false);

        v2f a1 = { (r10 < end && i10 - segBase == mn) ? 1.0f : 0.0f,
                   (r11 < end && i11 - segBase == mn) ? 1.0f : 0.0f };
        v2f b1 = { b10, b11 };
        c1 = __builtin_amdgcn_wmma_f32_16x16x4_f32(
                false, a1, false, b1, (short)0, c1, false, false);
    }

    v8f c = c0 + c1;

    // C/D layout: VGPR v holds M = v + 8*half, N = mn. Output row = graph; [G, 256] concat.
    #pragma unroll
    for (int v = 0; v < 8; ++v) {
        int g = segBase + v + (half << 3);
        out[(size_t)g * 256 + colOff + col] = c[v];
    }
}

extern "C" void kernel_launch(void* const* d_in, const int* in_sizes, int n_in,
                              void* d_out, int out_size, void* d_ws, size_t ws_size,
                              hipStream_t stream) {
    const float* xa = (const float*)d_in[0];   // x_atoms  [N_ATOMS, 128] fp32
    const float* xf = (const float*)d_in[1];   // x_frags  [N_FRAGS, 128] fp32
    const int*   ba = (const int*)d_in[2];     // batch       [N_ATOMS] int32 (sorted)
    const int*   bf = (const int*)d_in[3];     // frag_batch  [N_FRAGS] int32 (sorted)
    const int na = in_sizes[2];
    const int nf = in_sizes[3];

    const int G = out_size / 256;              // NUM_GRAPHS (out is [G, 256])
    dim3 grid(G / 16, 2, 1);                   // y=0: atoms -> cols [0,128); y=1: frags -> [128,256)
    frag_pool_wmma_kernel<<<grid, 256, 0, stream>>>(xa, xf, ba, bf, na, nf, (float*)d_out);
}